// GraphAttentionEncoder_51883204936014
// MI455X (gfx1250) — compile-verified
//
#include <hip/hip_runtime.h>
#include <hip/hip_bf16.h>

typedef float v2f __attribute__((ext_vector_type(2)));
typedef float v4f __attribute__((ext_vector_type(4)));
typedef float v8f __attribute__((ext_vector_type(8)));

#define GAT_U 100
#define GAT_V 200
#define GAT_N 300
#define GAT_HD 16
#define GAT_H 8
#define GAT_B 16
#define GAT_E 128
#define NEG_INF_E (-9000000000000000.0f)
#define LRELU_ALPHA 0.01f

__device__ __forceinline__ v8f wmma_f32_16x16x4(v2f a, v2f b, v8f c) {
  // V_WMMA_F32_16X16X4_F32: D = A(16x4 f32) x B(4x16 f32) + C(16x16 f32)
  return __builtin_amdgcn_wmma_f32_16x16x4_f32(false, a, false, b, (short)0, c, false, false);
}

__device__ __forceinline__ float leaky01(float x) { return x >= 0.0f ? x : LRELU_ALPHA * x; }
__device__ __forceinline__ float elu1(float x) { return x > 0.0f ? x : (__expf(x) - 1.0f); }

// ---------------------------------------------------------------------------
// Kernel 1: one GAT head per workgroup. grid = B*H = 128 blocks, 256 threads.
// h_in/h_out are [B*H, 300, 16] contiguous chunks (the raw reshape of x).
// ---------------------------------------------------------------------------
__global__ __launch_bounds__(256) void gat_layer_kernel(
    const float* __restrict__ h_in, const unsigned char* __restrict__ adj,
    const float* __restrict__ Wl, const float* __restrict__ al,
    float* __restrict__ h_out)
{
  __shared__ float sH[304 * 16];      // padded input rows (300 -> 304)
  __shared__ float sWh[304 * 16];     // Wh = h @ W^T
  __shared__ float sWt[16 * 16];      // W transposed: sWt[k*16+j] = W[j][k]
  __shared__ float sA1[16], sA2[16];
  __shared__ float sEu[112], sEv[208], sEs[304];
  __shared__ float sMU[112], sZU[112], sMV[208], sZV[208];
  __shared__ unsigned int sAdj[112 * 7];  // adjacency bits: row u, bit v

  const int bh   = blockIdx.x;
  const int b    = bh / GAT_H;
  const int head = bh % GAT_H;
  const int tid  = threadIdx.x;
  const int lane = tid & 31;
  const int wave = tid >> 5;
  const int lm    = lane & 15;
  const int khalf = (lane < 16) ? 0 : 2;   // K split across lane halves (A & B operands)
  const int mhalf = (lane < 16) ? 0 : 8;   // M split for C/D rows

  // ---- phase 1: stage inputs into LDS ----
  const float* hbase = h_in + bh * (GAT_N * GAT_HD);
  for (int i = tid; i < 304 * 16; i += 256)
    sH[i] = (i < GAT_N * GAT_HD) ? hbase[i] : 0.0f;
  { int j = tid >> 4, k = tid & 15; sWt[k * 16 + j] = Wl[head * 256 + j * 16 + k]; }
  if (tid < 16) { sA1[tid] = al[head * 32 + tid]; sA2[tid] = al[head * 32 + 16 + tid]; }
  const unsigned char* adj_b = adj + b * (GAT_U * GAT_V);
  for (int w = tid; w < 112 * 7; w += 256) {
    int u = w / 7, wo = w % 7;
    unsigned int bits = 0u;
    if (u < GAT_U) {
      int vb = wo * 32;
      int nlim = GAT_V - vb; if (nlim > 32) nlim = 32;
      const unsigned char* p = adj_b + u * GAT_V + vb;
      for (int j = 0; j < nlim; ++j) if (p[j]) bits |= (1u << j);
    }
    sAdj[w] = bits;
  }
  __syncthreads();

  // ---- phase 2: Wh = h @ W^T  (19 M-tiles x 4 K-steps of WMMA f32 16x16x4) ----
  for (int mt = wave; mt < 19; mt += 8) {
    v8f acc = {};
    const int mrow = mt * 16 + lm;
    #pragma unroll
    for (int ks = 0; ks < 4; ++ks) {
      int k0 = ks * 4 + khalf;
      v2f A, Bv;
      A.x = sH[mrow * 16 + k0];
      A.y = sH[mrow * 16 + k0 + 1];
      Bv.x = sWt[k0 * 16 + lm];
      Bv.y = sWt[(k0 + 1) * 16 + lm];
      acc = wmma_f32_16x16x4(A, Bv, acc);
    }
    const int rbase = mt * 16 + mhalf;
    #pragma unroll
    for (int r = 0; r < 8; ++r) sWh[(rbase + r) * 16 + lm] = acc[r];
  }
  __syncthreads();

  // ---- phase 3: eu = Wh[:100]@a1, ev = Wh[100:]@a2, e_self = leaky(Wh@(a1+a2)) ----
  for (int n = tid; n < 304; n += 256) {
    float d1 = 0.0f, d2 = 0.0f;
    #pragma unroll
    for (int k = 0; k < 16; ++k) { float w = sWh[n * 16 + k]; d1 += w * sA1[k]; d2 += w * sA2[k]; }
    sEs[n] = leaky01(d1 + d2);
    if (n < GAT_U) sEu[n] = d1;
    if (n >= GAT_U && n < GAT_N) sEv[n - GAT_U] = d2;
  }
  if (tid >= 100 && tid < 112) sEu[tid] = 0.0f;
  if (tid >= 200 && tid < 208) sEv[tid] = 0.0f;
  __syncthreads();

  // ---- phase 4: softmax stats (row-wise for attU, col-wise for attV) ----
  if (tid < GAT_U) {
    const int u = tid;
    const float eu_l = sEu[u];
    const unsigned int* ab = &sAdj[u * 7];
    float m = sEs[u];
    for (int v = 0; v < GAT_V; ++v) {
      float e = ((ab[v >> 5] >> (v & 31)) & 1u) ? NEG_INF_E : leaky01(eu_l + sEv[v]);
      m = fmaxf(m, e);
    }
    float z = __expf(sEs[u] - m);
    for (int v = 0; v < GAT_V; ++v) {
      float e = ((ab[v >> 5] >> (v & 31)) & 1u) ? NEG_INF_E : leaky01(eu_l + sEv[v]);
      z += __expf(e - m);
    }
    sMU[u] = m; sZU[u] = z;
  } else {
    for (int v = tid - 100; v < GAT_V; v += 156) {
      const float ev_l = sEv[v];
      const int wq = v >> 5; const unsigned int bm = 1u << (v & 31);
      float m = sEs[GAT_U + v];
      for (int u = 0; u < GAT_U; ++u) {
        float e = (sAdj[u * 7 + wq] & bm) ? NEG_INF_E : leaky01(sEu[u] + ev_l);
        m = fmaxf(m, e);
      }
      float z = __expf(sEs[GAT_U + v] - m);
      for (int u = 0; u < GAT_U; ++u) {
        float e = (sAdj[u * 7 + wq] & bm) ? NEG_INF_E : leaky01(sEu[u] + ev_l);
        z += __expf(e - m);
      }
      sMV[v] = m; sZV[v] = z;
    }
  }
  if (tid < 12)               { sMU[100 + tid] = 0.0f; sZU[100 + tid] = 1.0f; }
  if (tid >= 16 && tid < 24)  { sMV[184 + tid] = 0.0f; sZV[184 + tid] = 1.0f; } // 200..207
  __syncthreads();

  // ---- phase 5: hu (7 tiles) & hv (13 tiles) fused-softmax GEMMs via WMMA ----
  float* obase = h_out + bh * (GAT_N * GAT_HD);
  for (int t = wave; t < 20; t += 8) {
    if (t < 7) {
      // hu[u,:] = ( sum_v exp(e[u,v]-m)*Wh[100+v,:] + exp(es[u]-m)*Wh[u,:] ) / Z
      const int mt = t;
      const int u0 = mt * 16 + lm;                 // < 112 (padded)
      const float eu_l = sEu[u0];
      const float mu_l = sMU[u0];
      const unsigned int* ab = &sAdj[u0 * 7];
      v8f acc = {};
      for (int ks = 0; ks < 50; ++ks) {            // K = 200
        int k0 = ks * 4 + khalf;
        v2f A, Bv;
        float e0 = ((ab[k0 >> 5] >> (k0 & 31)) & 1u) ? NEG_INF_E : leaky01(eu_l + sEv[k0]);
        float e1 = ((ab[(k0 + 1) >> 5] >> ((k0 + 1) & 31)) & 1u) ? NEG_INF_E : leaky01(eu_l + sEv[k0 + 1]);
        A.x = __expf(e0 - mu_l);
        A.y = __expf(e1 - mu_l);
        Bv.x = sWh[(GAT_U + k0) * 16 + lm];
        Bv.y = sWh[(GAT_U + k0 + 1) * 16 + lm];
        acc = wmma_f32_16x16x4(A, Bv, acc);
      }
      const int rbase = mt * 16 + mhalf;
      #pragma unroll
      for (int r = 0; r < 8; ++r) {
        int u = rbase + r;
        if (u < GAT_U) {
          float val = acc[r] + __expf(sEs[u] - sMU[u]) * sWh[u * 16 + lm];
          obase[u * 16 + lm] = elu1(val / sZU[u]);
        }
      }
    } else {
      // hv[v,:] = ( sum_u exp(e[u,v]-m)*Wh[u,:] + exp(es[100+v]-m)*Wh[100+v,:] ) / Z
      const int mt = t - 7;
      const int v0 = mt * 16 + lm;                 // < 208 (padded)
      const float ev_l = sEv[v0];
      const float mv_l = sMV[v0];
      const int wq = v0 >> 5; const unsigned int bm = 1u << (v0 & 31);
      v8f acc = {};
      for (int ks = 0; ks < 25; ++ks) {            // K = 100
        int k0 = ks * 4 + khalf;
        v2f A, Bv;
        float e0 = (sAdj[k0 * 7 + wq] & bm) ? NEG_INF_E : leaky01(sEu[k0] + ev_l);
        float e1 = (sAdj[(k0 + 1) * 7 + wq] & bm) ? NEG_INF_E : leaky01(sEu[k0 + 1] + ev_l);
        A.x = __expf(e0 - mv_l);
        A.y = __expf(e1 - mv_l);
        Bv.x = sWh[k0 * 16 + lm];
        Bv.y = sWh[(k0 + 1) * 16 + lm];
        acc = wmma_f32_16x16x4(A, Bv, acc);
      }
      const int rbase = mt * 16 + mhalf;
      #pragma unroll
      for (int r = 0; r < 8; ++r) {
        int v = rbase + r;
        if (v < GAT_V) {
          float val = acc[r] + __expf(sEs[GAT_U + v] - sMV[v]) * sWh[(GAT_U + v) * 16 + lm];
          obase[(GAT_U + v) * 16 + lm] = elu1(val / sZV[v]);
        }
      }
    }
  }
}

// ---------------------------------------------------------------------------
// Kernel 2: pu = hE[:, :100] @ W1^T, pv = hE[:, 100:] @ W2^T  via WMMA.
// grid = B * (7 + 13) blocks; 8 waves = 8 N-tiles of 16 columns.
// ---------------------------------------------------------------------------
__global__ __launch_bounds__(256) void pupv_kernel(
    const float* __restrict__ hE, const float* __restrict__ W_last,
    float* __restrict__ pu, float* __restrict__ pv)
{
  const int blk = blockIdx.x;
  const int b = blk / 20;
  const int t = blk % 20;
  const int lane = threadIdx.x & 31;
  const int wave = threadIdx.x >> 5;           // N-tile index 0..7
  const int lm = lane & 15;
  const int khalf = (lane < 16) ? 0 : 2;
  const int mhalf = (lane < 16) ? 0 : 8;

  const bool isPu = (t < 7);
  const int mt = isPu ? t : t - 7;
  const int rowbase = isPu ? mt * 16 : GAT_U + mt * 16;
  const int wofs = isPu ? 0 : GAT_E;           // W1 vs W2 column offset in W_last
  const int j = wave * 16 + lm;                // output column

  const float* Arow = hE + (b * GAT_N + rowbase + lm) * GAT_E;  // A: M=lane&15
  const float* Brow = W_last + j * 257 + wofs;                  // B[k][j] = W[j][k]

  v8f acc = {};
  for (int ks = 0; ks < 32; ++ks) {            // K = 128
    int k0 = ks * 4 + khalf;
    v2f A, Bv;
    A.x = Arow[k0];     A.y = Arow[k0 + 1];
    Bv.x = Brow[k0];    Bv.y = Brow[k0 + 1];
    acc = wmma_f32_16x16x4(A, Bv, acc);
  }
  const int r0 = mt * 16 + mhalf;
  const int lim = isPu ? GAT_U : GAT_V;
  float* outp = isPu ? (pu + b * GAT_U * GAT_E) : (pv + b * GAT_V * GAT_E);
  #pragma unroll
  for (int r = 0; r < 8; ++r) {
    int row = r0 + r;
    if (row < lim) outp[row * GAT_E + j] = acc[r];
  }
}

// ---------------------------------------------------------------------------
// Kernel 3: the bandwidth kernel. out[b,u,v,:] = mask*(pu[b,u,:]+pv[b,v,:]) + w*wl
// 163.8 MB of non-temporal 128-bit stores. grid = B*U = 1600 blocks.
// ---------------------------------------------------------------------------
__global__ __launch_bounds__(256) void out_kernel(
    const float* __restrict__ pu, const float* __restrict__ pv,
    const unsigned char* __restrict__ adj, const float* __restrict__ weights,
    const float* __restrict__ W_last, float* __restrict__ out)
{
  __shared__ float sPu[GAT_E], sWl[GAT_E];
  const int blk = blockIdx.x;
  const int b = blk / GAT_U;
  const int u = blk % GAT_U;
  const int tid = threadIdx.x;
  if (tid < GAT_E) {
    sPu[tid] = pu[(b * GAT_U + u) * GAT_E + tid];
    sWl[tid] = W_last[tid * 257 + 2 * GAT_E];   // last column of W_last
  }
  __syncthreads();

  const int e4 = (tid & 31) * 4;
  const int vg = tid >> 5;                      // 8 v-rows in flight
  const v4f pu4 = *(const v4f*)&sPu[e4];
  const v4f wl4 = *(const v4f*)&sWl[e4];
  const unsigned char* arow = adj + (b * GAT_U + u) * GAT_V;
  const float* wrow = weights + (b * GAT_U + u) * GAT_V;
  float* orow = out + (size_t)(b * GAT_U * GAT_V + u * GAT_V) * GAT_E;

  for (int v = vg; v < GAT_V; v += 8) {
    float m = arow[v] ? 0.0f : 1.0f;            // mask = 1 - adj
    float w = wrow[v];
    v4f pv4 = *(const v4f*)(pv + ((size_t)(b * GAT_V + v)) * GAT_E + e4);
    v4f o = m * (pu4 + pv4) + w * wl4;
    __builtin_nontemporal_store(o, (v4f*)(orow + (size_t)v * GAT_E + e4));
  }
}

// ---------------------------------------------------------------------------
extern "C" void kernel_launch(void* const* d_in, const int* in_sizes, int n_in,
                              void* d_out, int out_size, void* d_ws, size_t ws_size,
                              hipStream_t stream) {
  (void)in_sizes; (void)n_in; (void)out_size; (void)ws_size;
  const float* x               = (const float*)d_in[0];
  const unsigned char* adj     = (const unsigned char*)d_in[1];   // bool [B,U,V]
  const float* weights         = (const float*)d_in[2];
  const float* W_layers        = (const float*)d_in[3];           // [3,8,16,16]
  const float* a_layers        = (const float*)d_in[4];           // [3,8,32]
  const float* W_last          = (const float*)d_in[5];           // [128,257]
  float* out = (float*)d_out;

  float* ws = (float*)d_ws;
  const int HCH = GAT_B * GAT_H * GAT_N * GAT_HD;   // 614400 floats per h buffer
  float* h0 = ws;
  float* h1 = h0 + HCH;
  float* pu = h1 + HCH;                              // [B,100,128]
  float* pv = pu + GAT_B * GAT_U * GAT_E;            // [B,200,128]

  const int WSTRIDE = GAT_H * GAT_HD * GAT_HD;       // 2048 per layer
  const int ASTRIDE = GAT_H * 2 * GAT_HD;            // 256 per layer

  gat_layer_kernel<<<GAT_B * GAT_H, 256, 0, stream>>>(x,  adj, W_layers + 0 * WSTRIDE, a_layers + 0 * ASTRIDE, h0);
  gat_layer_kernel<<<GAT_B * GAT_H, 256, 0, stream>>>(h0, adj, W_layers + 1 * WSTRIDE, a_layers + 1 * ASTRIDE, h1);
  gat_layer_kernel<<<GAT_B * GAT_H, 256, 0, stream>>>(h1, adj, W_layers + 2 * WSTRIDE, a_layers + 2 * ASTRIDE, h0);
  pupv_kernel<<<GAT_B * 20, 256, 0, stream>>>(h0, W_last, pu, pv);
  out_kernel<<<GAT_B * GAT_U, 256, 0, stream>>>(pu, pv, adj, weights, W_last, out);
}